// LinearRationalSpline_6133213298950
// MI455X (gfx1250) — compile-verified
//
#include <hip/hip_runtime.h>
#include <hip/hip_bf16.h>
#include <math.h>

// Linear Rational Spline forward for MI455X (gfx1250, wave32).
// Memory-bound: ~268MB read / 8MB write -> ~12us floor @ 23.3 TB/s.
// Strategy: stage each block's 256x63 fp32 params into LDS via CDNA5
// async global->LDS DMA (ASYNCcnt path), conflict-free LDS reads
// (stride 63 vs 64 banks), fully fused per-thread spline evaluation.

#define NBINS 16
#define NPAR  63          // 4*K-1 params per element
#define TPB   256

__device__ __forceinline__ float softplus_f(float v) {
    // numerically stable: max(v,0) + log1p(exp(-|v|))
    return fmaxf(v, 0.0f) + log1pf(__expf(-fabsf(v)));
}

__global__ __launch_bounds__(TPB) void lrs_kernel(
    const float* __restrict__ inputs,
    const float* __restrict__ params,
    float* __restrict__ out,      // [N] outputs, then [N] logabsdet
    int N)
{
    __shared__ float sp[TPB * NPAR];   // 64512 bytes

    const int tid = threadIdx.x;
    const int blockBase = blockIdx.x * TPB;

    // ---- Stage this block's params into LDS with async DMA ----
    int elems = N - blockBase;
    if (elems > TPB) elems = TPB;
    const int totalFloats = elems * NPAR;       // 16128 for a full block
    const int nF4 = totalFloats >> 2;           // 4032 float4 chunks
    const int remBase = nF4 << 2;

    const uint64_t gbase  = (uint64_t)(uintptr_t)(params + (size_t)blockBase * NPAR);
    const uint32_t ldsBase = (uint32_t)(uintptr_t)(&sp[0]); // wave-relative LDS byte addr

    for (int i = tid; i < nF4; i += TPB) {
        uint32_t off = (uint32_t)i << 4;
        uint32_t la  = ldsBase + off;
        asm volatile("global_load_async_to_lds_b128 %0, %1, %2"
                     :: "v"(la), "v"(off), "s"(gbase) : "memory");
    }
    if (tid < (totalFloats - remBase)) {         // up to 3 tail floats
        uint32_t off = (uint32_t)((remBase + tid) << 2);
        uint32_t la  = ldsBase + off;
        asm volatile("global_load_async_to_lds_b32 %0, %1, %2"
                     :: "v"(la), "v"(off), "s"(gbase) : "memory");
    }
    asm volatile("s_wait_asynccnt 0x0" ::: "memory");
    __syncthreads();

    const int e = blockBase + tid;
    if (e >= N) return;

    const float x = inputs[e];
    float* __restrict__ p = &sp[tid * NPAR];   // stride 63: bank-conflict-free

    // ---------- widths: softmax -> knots -> bin search ----------
    float mw = p[0];
#pragma unroll
    for (int j = 1; j < NBINS; ++j) mw = fmaxf(mw, p[j]);
    float sw = 0.0f;
#pragma unroll
    for (int j = 0; j < NBINS; ++j) { float t = __expf(p[j] - mw); p[j] = t; sw += t; }
    const float invw = __frcp_rn(sw);

    const float WSCALE = 1.0f - 0.001f * 16.0f;   // (1 - MIN_BIN_WIDTH*K)
    float cum = 0.0f, cPrev = -3.0f;
    int idx = 0;
    float cwSel = -3.0f, wSel = 0.0f;
#pragma unroll
    for (int jj = 0; jj < NBINS; ++jj) {
        float wj = 0.001f + WSCALE * (p[jj] * invw);
        cum += wj;
        float cCur = (jj == NBINS - 1) ? 3.0f : fmaf(cum, 6.0f, -3.0f);
        // searchsorted(cumwidths+eps, x, right) - 1, clipped to [0,15]
        bool take = (jj == 0) || (cPrev + 1e-6f <= x);
        if (take) { idx = jj; cwSel = cPrev; wSel = cCur - cPrev; }
        cPrev = cCur;
    }

    // ---------- heights: softmax -> knots, select bin idx ----------
    float* ph = p + NBINS;
    float mh = ph[0];
#pragma unroll
    for (int j = 1; j < NBINS; ++j) mh = fmaxf(mh, ph[j]);
    float sh = 0.0f;
#pragma unroll
    for (int j = 0; j < NBINS; ++j) { float t = __expf(ph[j] - mh); ph[j] = t; sh += t; }
    const float invh = __frcp_rn(sh);

    float cumh = 0.0f, chPrev = -3.0f;
    float chSel = -3.0f, hSel = 0.0f;
#pragma unroll
    for (int jj = 0; jj < NBINS; ++jj) {
        float hj = 0.001f + WSCALE * (ph[jj] * invh);
        cumh += hj;
        float chCur = (jj == NBINS - 1) ? 3.0f : fmaf(cumh, 6.0f, -3.0f);
        if (jj == idx) { chSel = chPrev; hSel = chCur - chPrev; }
        chPrev = chCur;
    }

    // ---------- derivatives (offset 32, 15 values; edges = 1-MIN_DERIVATIVE) ----------
    const float EDGE = 1.0f - 0.001f;
    float dlo = (idx == 0)          ? EDGE : (0.001f + softplus_f(p[32 + idx - 1]));
    float dhi = (idx == NBINS - 1)  ? EDGE : (0.001f + softplus_f(p[32 + idx]));

    // ---------- lambda (offset 47): sigmoid then affine ----------
    float lraw = p[47 + idx];
    float lam = fmaf(0.95f, __frcp_rn(1.0f + __expf(-lraw)), 0.025f);

    // ---------- rational spline evaluation ----------
    float delta = hSel / wSel;
    float wb  = sqrtf(dlo / dhi);
    float lwb = lam * wb;
    float wc  = (lam * dlo + (wb - lwb) * dhi) / delta;
    float ya  = chSel;
    float yb  = hSel + chSel;
    float l1  = 1.0f - lam;
    float yc  = (lwb * yb + l1 * ya) / (l1 + lwb);

    float theta = (x - cwSel) / wSel;
    bool  ind = (theta <= lam);
    float ltheta = lam - theta;
    float wcyc = wc * yc;
    float wcyctheta = wcyc * theta;
    float num = ind ? (wcyctheta + ya * ltheta)
                    : ((wcyc - wcyctheta) - wb * yb * ltheta);
    float wctheta = wc * theta;
    float den = ind ? (wctheta + ltheta)
                    : ((wc - wctheta) - wb * ltheta);
    float outv = num / den;
    float dnum = wc * (ind ? (lam * (yc - ya)) : ((wb - lwb) * (yb - yc))) / wSel;
    float lad  = __logf(dnum) - 2.0f * __logf(fabsf(den));

    bool outside = (x < -3.0f) || (x > 3.0f);
    outv = outside ? x    : outv;
    lad  = outside ? 0.0f : lad;

    out[e]     = outv;
    out[N + e] = lad;
}

extern "C" void kernel_launch(void* const* d_in, const int* in_sizes, int n_in,
                              void* d_out, int out_size, void* d_ws, size_t ws_size,
                              hipStream_t stream) {
    const float* inputs = (const float*)d_in[0];
    const float* params = (const float*)d_in[1];
    float* out = (float*)d_out;
    int N = in_sizes[0];                 // 16384 * 64 = 1,048,576 elements
    int blocks = (N + TPB - 1) / TPB;
    lrs_kernel<<<blocks, TPB, 0, stream>>>(inputs, params, out, N);
}